// ModelNew_50388556316931
// MI455X (gfx1250) — compile-verified
//
#include <hip/hip_runtime.h>
#include <hip/hip_bf16.h>

// ConvTranspose2d(64,64,3,stride=1,pad=0,bias=False) on [1,64,1024,1024] f32
// == 3x3 full correlation (flipped kernel) over input zero-padded by 2.
// Implicit GEMM on V_WMMA_F32_16X16X4_F32 (wave32):
//   D[co,px] += A[co,ci] * B[ci,px]   (M=co so NCHW stores are contiguous)
//
// 512 threads = 16 waves per WG; each wave: 2 co-groups x 4 pixel-groups
// -> 64 accumulator VGPRs, total pressure < 256 (no VGPR-MSB mode churn,
// higher occupancy for the HBM-bound stream).

#define HIN   1024
#define WIN   1024
#define HOUT  1026
#define WOUT  1026
#define CI    64
#define CO    64
#define TH    16          // output rows per workgroup
#define TW    32          // output cols per workgroup
#define CIT   16          // ci channels staged in LDS per chunk
#define LDSH  18          // TH + 2
#define LDSW  36          // TW + 2 (=34) padded to 36 floats
#define NTHREADS 512

typedef __attribute__((ext_vector_type(2))) float v2f;
typedef __attribute__((ext_vector_type(8))) float v8f;

__launch_bounds__(NTHREADS)
__global__ void convtrans_wmma_f32(const float* __restrict__ x,
                                   const float* __restrict__ w,
                                   float* __restrict__ out) {
    // 16 ci-slices of an 18x36 activation tile: 41,472 B of LDS
    __shared__ float xs[CIT * LDSH * LDSW];

    const int tid  = threadIdx.x;
    const int wave = tid >> 5;        // 16 waves
    const int lane = tid & 31;
    const int half = lane >> 4;       // 0: lanes 0-15, 1: lanes 16-31
    const int m    = lane & 15;

    const int cghalf = wave & 1;      // owns co-groups 2*cghalf, 2*cghalf+1
    const int pgbase = (wave >> 1) * 4;  // owns pixel-groups pgbase..pgbase+3

    const int ox0 = blockIdx.x * TW;
    const int oy0 = blockIdx.y * TH;

    v8f acc[2][4];   // [cg][pg]
#pragma unroll
    for (int i = 0; i < 2; ++i)
#pragma unroll
        for (int j = 0; j < 4; ++j)
#pragma unroll
            for (int r = 0; r < 8; ++r) acc[i][j][r] = 0.0f;

    for (int ci0 = 0; ci0 < CI; ci0 += CIT) {
        __syncthreads();  // protect LDS reads of previous chunk
        // ---- stage input region [oy0-2, oy0+15] x [ox0-2, ox0+31], zero-padded
        for (int idx = tid; idx < CIT * LDSH * 34; idx += NTHREADS) {
            int c    = idx % 34;
            int rest = idx / 34;
            int r    = rest % LDSH;
            int cl   = rest / LDSH;
            int iy   = oy0 - 2 + r;
            int ix   = ox0 - 2 + c;
            float v  = 0.0f;
            if ((unsigned)iy < (unsigned)HIN && (unsigned)ix < (unsigned)WIN)
                v = x[(size_t)(ci0 + cl) * (HIN * WIN) + (size_t)iy * WIN + ix];
            xs[(cl * LDSH + r) * LDSW + c] = v;
        }
        __syncthreads();

        // ---- software prefetch of next ci-chunk's region (global_prefetch_b8)
        if (ci0 + CIT < CI) {
            for (int idx = tid; idx < CIT * LDSH * 3; idx += NTHREADS) {
                int seg  = idx % 3;               // one 64B line per 16 floats
                int rest = idx / 3;
                int r    = rest % LDSH;
                int cl   = rest / LDSH;
                int iy   = oy0 - 2 + r;
                int ix   = ox0 - 2 + seg * 16;
                if ((unsigned)iy < (unsigned)HIN && (unsigned)ix < (unsigned)WIN)
                    __builtin_prefetch(
                        &x[(size_t)(ci0 + CIT + cl) * (HIN * WIN) +
                           (size_t)iy * WIN + ix], 0, 3);
            }
        }

        // ---- K loop: 9 taps x 4 sub-chunks of 4 ci
        for (int ky = 0; ky < 3; ++ky) {
            for (int kx = 0; kx < 3; ++kx) {
                const int tap = ky * 3 + kx;
#pragma unroll
                for (int kc = 0; kc < 4; ++kc) {
                    const int cb = kc * 4 + 2 * half;  // lane's ci rows: cb, cb+1

                    // A fragments (16 co x 4 ci): w[ci][co][tap], gathered from
                    // global (144 KB total, resident in WGP$/L2)
                    v2f A[2];
#pragma unroll
                    for (int cg = 0; cg < 2; ++cg) {
                        const int co = (2 * cghalf + cg) * 16 + m;
                        const float* wp =
                            w + ((size_t)(ci0 + cb) * CO + co) * 9 + tap;
                        A[cg].x = wp[0];           // ci = ci0+cb
                        A[cg].y = wp[CO * 9];      // ci = ci0+cb+1
                    }

                    // B fragments (4 ci x 16 px) from LDS, always in-range
                    v2f B[4];
#pragma unroll
                    for (int pg = 0; pg < 4; ++pg) {
                        const int g   = pgbase + pg;     // 0..31 pixel groups
                        const int ty  = g >> 1;          // tile row
                        const int txb = (g & 1) * 16;    // tile col base
                        const int a0  = (cb * LDSH + ty + 2 - ky) * LDSW
                                        + txb + m + 2 - kx;
                        B[pg].x = xs[a0];
                        B[pg].y = xs[a0 + LDSH * LDSW];  // next ci slice
                    }

#pragma unroll
                    for (int cg = 0; cg < 2; ++cg)
#pragma unroll
                        for (int pg = 0; pg < 4; ++pg)
                            acc[cg][pg] = __builtin_amdgcn_wmma_f32_16x16x4_f32(
                                false, A[cg], false, B[pg],
                                (short)0, acc[cg][pg], false, false);
                }
            }
        }
    }

    // ---- store: D VGPR r -> co = cg*16 + r + 8*half, px = m (64B runs)
#pragma unroll
    for (int cg = 0; cg < 2; ++cg) {
#pragma unroll
        for (int pg = 0; pg < 4; ++pg) {
            const int g  = pgbase + pg;
            const int oy = oy0 + (g >> 1);
            const int ox = ox0 + (g & 1) * 16 + m;
            if (oy < HOUT && ox < WOUT) {
#pragma unroll
                for (int r = 0; r < 8; ++r) {
                    const int co = (2 * cghalf + cg) * 16 + r + 8 * half;
                    out[(size_t)co * (HOUT * WOUT) + (size_t)oy * WOUT + ox] =
                        acc[cg][pg][r];
                }
            }
        }
    }
}

extern "C" void kernel_launch(void* const* d_in, const int* in_sizes, int n_in,
                              void* d_out, int out_size, void* d_ws, size_t ws_size,
                              hipStream_t stream) {
    const float* x = (const float*)d_in[0];  // [64,1024,1024]
    const float* w = (const float*)d_in[1];  // [64,64,3,3]
    float* out = (float*)d_out;              // [64,1026,1026]

    dim3 grid((WOUT + TW - 1) / TW, (HOUT + TH - 1) / TH);  // 33 x 65
    dim3 block(NTHREADS);
    convtrans_wmma_f32<<<grid, block, 0, stream>>>(x, w, out);
}